// EasyRec_58454504898835
// MI455X (gfx1250) — compile-verified
//
#include <hip/hip_runtime.h>

typedef __attribute__((ext_vector_type(2))) float v2f;
typedef __attribute__((ext_vector_type(8))) float v8f;

static constexpr int Dd  = 256;   // embedding dim D
static constexpr int K2  = 512;   // GEMM reduction dim = 2*D
static constexpr int KN  = 64;    // max neighbors K
static constexpr int TM  = 16;    // batch rows per workgroup
static constexpr int LDA = 516;   // padded LDS row stride in floats (516 % 64 == 4 -> conflict-free)

// ---- One-shot repack of W [256][512] row-major into WMMA-fragment-native layout.
// Wp (as v2f): Wp[(kk/4)*512 + n*2 + h] = { W[n][kk+2h], W[n][kk+2h+1] }
// so a wave's B-fragment load (lane = h*16 + l15, col n = n0 + l15) is one b64 per lane
// covering a fully contiguous 256-byte run across the 32 lanes.
__global__ __launch_bounds__(256) void wt_pack(const float* __restrict__ W,
                                               v2f* __restrict__ Wp) {
  int idx = blockIdx.x * 256 + threadIdx.x;   // 65536 v2f elements, grid = 256
  int kkg = idx >> 9;                         // K-step group (kk/4), 128 groups
  int rem = idx & 511;                        // n*2 + h
  int n   = rem >> 1;
  int h   = rem & 1;
  const float* src = W + n * K2 + kkg * 4 + 2 * h;
  v2f p;
  p.x = src[0];
  p.y = src[1];
  Wp[idx] = p;
}

// ---- Fully fused: gather-mean -> cat tile in LDS -> WMMA fp32 GEMM -> bias/select -> v2e dot ----
template <bool USE_WP>
__global__ __launch_bounds__(256) void easyrec_fused(
    const int*   __restrict__ nodes_u,
    const int*   __restrict__ nodes_v,
    const int*   __restrict__ neighbors,
    const int*   __restrict__ ncnt,
    const float* __restrict__ u2e,
    const float* __restrict__ v2e,
    const float* __restrict__ Wsrc,   // packed Wp if USE_WP, else raw W [256][512]
    const float* __restrict__ bias,
    float*       __restrict__ out) {
  __shared__ float a_lds[TM * LDA];   // cat tile: [row][0:256]=self, [row][256:512]=nmean
  __shared__ float part[8][TM];       // per-wave row partial dot products

  const int tid  = threadIdx.x;
  const int lane = tid & 31;
  const int w    = tid >> 5;          // wave id 0..7
  const int b0   = blockIdx.x * TM;

  // ---------------- Phase 1: gather self + mean(neighbors) into LDS ----------------
  // Wave w owns batch rows 2w and 2w+1; each lane handles 8 contiguous columns.
  #pragma unroll
  for (int rr = 0; rr < 2; ++rr) {
    const int r   = w * 2 + rr;
    const int b   = b0 + r;
    const int nu  = nodes_u[b];
    const int cnt = ncnt[b];

    const float4* up = (const float4*)(u2e + (size_t)nu * Dd);
    float4 s0 = up[lane * 2];
    float4 s1 = up[lane * 2 + 1];
    *(float4*)&a_lds[r * LDA + lane * 8]     = s0;
    *(float4*)&a_lds[r * LDA + lane * 8 + 4] = s1;

    float4 a0 = make_float4(0.f, 0.f, 0.f, 0.f);
    float4 a1 = make_float4(0.f, 0.f, 0.f, 0.f);
    const int* nb = neighbors + (size_t)b * KN;
    #pragma unroll 4
    for (int k = 0; k < cnt; ++k) {
      const float4* np = (const float4*)(u2e + (size_t)nb[k] * Dd);
      float4 t0 = np[lane * 2];
      float4 t1 = np[lane * 2 + 1];
      a0.x += t0.x; a0.y += t0.y; a0.z += t0.z; a0.w += t0.w;
      a1.x += t1.x; a1.y += t1.y; a1.z += t1.z; a1.w += t1.w;
    }
    const float inv = 1.0f / (float)(cnt > 0 ? cnt : 1);
    a0.x *= inv; a0.y *= inv; a0.z *= inv; a0.w *= inv;
    a1.x *= inv; a1.y *= inv; a1.z *= inv; a1.w *= inv;
    *(float4*)&a_lds[r * LDA + Dd + lane * 8]     = a0;
    *(float4*)&a_lds[r * LDA + Dd + lane * 8 + 4] = a1;
  }
  __syncthreads();

  // ---------------- Phase 2: proj = cat @ W^T via V_WMMA_F32_16X16X4_F32 ----------------
  // Wave w computes output cols [w*32, w*32+16) and [w*32+16, w*32+32) for rows b0..b0+15.
  const int half = lane >> 4;     // K-half selector within fragments
  const int l15  = lane & 15;
  const int n0   = w * 32;
  const int aBase = l15 * LDA + half * 2;  // dword index: A[l15][kk + 2*half .. +1]

  v8f acc0 = {};
  v8f acc1 = {};

  if (USE_WP) {
    // Fragment-native packed W: one contiguous b64 per lane per fragment.
    const v2f* wp = (const v2f*)Wsrc;
    const int wBase = (n0 + l15) * 2 + half;
    #pragma unroll 4
    for (int kk = 0; kk < K2; kk += 4) {
      v2f a = *(const v2f*)&a_lds[aBase + kk];
      const int wi = wBase + (kk >> 2) * 512;
      v2f bA = wp[wi];
      v2f bB = wp[wi + 32];
      acc0 = __builtin_amdgcn_wmma_f32_16x16x4_f32(false, a, false, bA, (short)0, acc0, false, false);
      acc1 = __builtin_amdgcn_wmma_f32_16x16x4_f32(false, a, false, bB, (short)0, acc1, false, false);
    }
  } else {
    // W[n][k] row-major fallback: per-lane contiguous b64 along k (uncoalesced across lanes)
    const int wRowA = (n0 + l15) * K2 + half * 2;
    const int wRowB = wRowA + 16 * K2;
    #pragma unroll 4
    for (int kk = 0; kk < K2; kk += 4) {
      v2f a  = *(const v2f*)&a_lds[aBase + kk];
      v2f bA = *(const v2f*)&Wsrc[wRowA + kk];
      v2f bB = *(const v2f*)&Wsrc[wRowB + kk];
      acc0 = __builtin_amdgcn_wmma_f32_16x16x4_f32(false, a, false, bA, (short)0, acc0, false, false);
      acc1 = __builtin_amdgcn_wmma_f32_16x16x4_f32(false, a, false, bB, (short)0, acc1, false, false);
    }
  }

  // ---------------- Phase 3: bias + (cnt>0 ? proj : self) + dot with v2e, reduce ----------------
  // D-matrix layout: VGPR v -> lanes 0-15: (M=v, N=l15), lanes 16-31: (M=v+8, N=l15).
  const float bias0 = bias[n0 + l15];
  const float bias1 = bias[n0 + 16 + l15];
  #pragma unroll
  for (int v = 0; v < 8; ++v) {
    const int row = half ? (v + 8) : v;
    const int b   = b0 + row;
    const int cnt = ncnt[b];
    const int nv  = nodes_v[b];
    float u0, u1;
    if (cnt > 0) {
      u0 = acc0[v] + bias0;
      u1 = acc1[v] + bias1;
    } else {
      u0 = a_lds[row * LDA + n0 + l15];        // raw self embedding
      u1 = a_lds[row * LDA + n0 + 16 + l15];
    }
    const float* vrow = v2e + (size_t)nv * Dd;
    float p = u0 * vrow[n0 + l15] + u1 * vrow[n0 + 16 + l15];
    // deterministic reduction over the 16 lanes of each half (xor masks stay in-half)
    p += __shfl_xor(p, 1);
    p += __shfl_xor(p, 2);
    p += __shfl_xor(p, 4);
    p += __shfl_xor(p, 8);
    if (l15 == 0) part[w][row] = p;
  }
  __syncthreads();

  if (tid < TM) {
    float s = 0.f;
    #pragma unroll
    for (int ww = 0; ww < 8; ++ww) s += part[ww][tid];
    out[b0 + tid] = s;
  }
}

extern "C" void kernel_launch(void* const* d_in, const int* in_sizes, int n_in,
                              void* d_out, int out_size, void* d_ws, size_t ws_size,
                              hipStream_t stream) {
  const int*   nodes_u   = (const int*)d_in[0];
  const int*   nodes_v   = (const int*)d_in[1];
  const int*   neighbors = (const int*)d_in[2];
  const int*   ncnt      = (const int*)d_in[3];
  const float* u2e       = (const float*)d_in[4];
  const float* v2e       = (const float*)d_in[5];
  const float* W         = (const float*)d_in[6];
  const float* bias      = (const float*)d_in[7];
  float*       out       = (float*)d_out;

  const int B = in_sizes[0];
  const int nblocks = B / TM;   // B = 16384 -> 1024 workgroups

  const size_t wpBytes = (size_t)K2 * Dd * sizeof(float);  // 512 KB
  if (d_ws != nullptr && ws_size >= wpBytes) {
    v2f* Wp = (v2f*)d_ws;
    wt_pack<<<(K2 / 4) * Dd * 2 / 256, 256, 0, stream>>>(W, Wp);
    easyrec_fused<true><<<nblocks, 256, 0, stream>>>(
        nodes_u, nodes_v, neighbors, ncnt, u2e, v2e, (const float*)Wp, bias, out);
  } else {
    easyrec_fused<false><<<nblocks, 256, 0, stream>>>(
        nodes_u, nodes_v, neighbors, ncnt, u2e, v2e, W, bias, out);
  }
}